// CRF_Layer_17652315586949
// MI455X (gfx1250) — compile-verified
//
#include <hip/hip_runtime.h>
#include <hip/hip_bf16.h>
#include <math.h>

typedef __attribute__((ext_vector_type(8))) float v8f;
typedef __attribute__((ext_vector_type(4))) float v4f;
typedef __attribute__((ext_vector_type(2))) float v2f;

#define BSZ 512
#define TLEN 1024
#define NLAB 64
#define NEGV -10000.0f

#if __has_builtin(__builtin_amdgcn_global_load_async_to_lds_b128) && \
    __has_builtin(__builtin_amdgcn_s_wait_asynccnt)
#define USE_ASYNC 1
typedef int v4i __attribute__((vector_size(16)));
typedef __attribute__((address_space(1))) v4i gv4i;   // global AS pointer pointee
typedef __attribute__((address_space(3))) v4i lv4i;   // LDS AS pointer pointee
#else
#define USE_ASYNC 0
#endif

__device__ __forceinline__ float fast_log(float x) {
#if __has_builtin(__builtin_amdgcn_logf)
    return __builtin_amdgcn_logf(x) * 0.6931471805599453f;  // v_log_f32 is log2
#else
    return __logf(x);
#endif
}

__launch_bounds__(128, 1)
__global__ void crf_fwd_kernel(const float* __restrict__ logits,
                               const int* __restrict__ lens,
                               const float* __restrict__ trans,
                               float* __restrict__ out) {
    // LDS state for a 16-batch tile
    __shared__ __align__(16) float alpha[16 * 64];     // running alpha
    __shared__ __align__(16) float pmat[16 * 64];      // exp(alpha - m)
    __shared__ __align__(16) float lt[2][16 * 64];     // double-buffered logit tile
    __shared__ float pm[16 * 8];                       // partial maxes
    __shared__ float mrow[16];                         // per-batch max
    __shared__ float ts[64];                           // trans[stop, :]
    __shared__ int   lens_s[16];

    const int tid  = threadIdx.x;
    const int lane = tid & 31;
    const int wave = tid >> 5;          // 4 waves; wave owns labels [16w,16w+16)
    const int b0   = blockIdx.x * 16;   // batch tile origin

    // ---- init alpha: NEG everywhere except start index (L-2) = 0
    for (int k = tid; k < 16 * 64; k += 128)
        alpha[k] = ((k & 63) == (NLAB - 2)) ? 0.0f : NEGV;
    if (tid < 64) ts[tid] = trans[(NLAB - 1) * 64 + tid];   // stop row
    if (tid < 16) lens_s[tid] = lens[b0 + tid];

    // ---- constant B fragments: B[k][n] = exp(trans[n0+n][k0+k])
    // 16x4 f32 striping: lanes 0-15 hold K=k0,k0+1; lanes 16-31 hold K=k0+2,k0+3
    const int ncol = wave * 16 + (lane & 15);
    const int koff = (lane >> 4) * 2;
    v2f bfrag[16];
    #pragma unroll
    for (int c = 0; c < 16; ++c) {
        v2f e = *(const v2f*)&trans[ncol * 64 + 4 * c + koff];
        bfrag[c].x = __expf(e.x);
        bfrag[c].y = __expf(e.y);
    }

    // ---- per-thread mapping for max/exp/logit phases: 128 threads = 16 b x 8 chunks
    const int br = tid >> 3;          // batch row 0..15
    const int j0 = (tid & 7) * 8;     // label chunk start

    const float* lrow = logits + (size_t)(b0 + br) * TLEN * 64 + j0;

#if USE_ASYNC
    // kick off async global->LDS copy of tile 0
    __builtin_amdgcn_global_load_async_to_lds_b128(
        (gv4i*)lrow, (lv4i*)&lt[0][br * 64 + j0], 0, 0);
    __builtin_amdgcn_global_load_async_to_lds_b128(
        (gv4i*)(lrow + 4), (lv4i*)&lt[0][br * 64 + j0 + 4], 0, 0);
#else
    v4f r0 = *(const v4f*)(lrow);
    v4f r1 = *(const v4f*)(lrow + 4);
#endif

    __syncthreads();

    const int arow = lane & 15;       // A-fragment batch row for this lane

    for (int t = 0; t < TLEN; ++t) {
        float* ltc = lt[t & 1];

#if !USE_ASYNC
        // stage prefetched logits into LDS (consumed after WMMA this step)
        *(v4f*)&ltc[br * 64 + j0]     = r0;
        *(v4f*)&ltc[br * 64 + j0 + 4] = r1;
#endif

        // phase A: per-batch max over labels
        float mx = alpha[br * 64 + j0];
        #pragma unroll
        for (int q = 1; q < 8; ++q) mx = fmaxf(mx, alpha[br * 64 + j0 + q]);
        pm[br * 8 + (tid & 7)] = mx;
        __syncthreads();
        if (tid < 16) {
            float m = pm[tid * 8];
            #pragma unroll
            for (int q = 1; q < 8; ++q) m = fmaxf(m, pm[tid * 8 + q]);
            mrow[tid] = m;
        }
        __syncthreads();

        // prefetch next step's logits into the other LDS buffer
        if (t + 1 < TLEN) {
            const float* lnx = lrow + (size_t)(t + 1) * 64;
#if USE_ASYNC
            float* ltn = lt[(t + 1) & 1];
            __builtin_amdgcn_global_load_async_to_lds_b128(
                (gv4i*)lnx, (lv4i*)&ltn[br * 64 + j0], 0, 0);
            __builtin_amdgcn_global_load_async_to_lds_b128(
                (gv4i*)(lnx + 4), (lv4i*)&ltn[br * 64 + j0 + 4], 0, 0);
#else
            r0 = *(const v4f*)(lnx);
            r1 = *(const v4f*)(lnx + 4);
#endif
        }

        // phase B: p = exp(alpha - m)
        {
            float m = mrow[br];
            #pragma unroll
            for (int q = 0; q < 8; ++q)
                pmat[br * 64 + j0 + q] = __expf(alpha[br * 64 + j0 + q] - m);
        }

#if USE_ASYNC
        // tile-t's 2 async loads are the oldest outstanding; FIFO order means
        // waiting down to <=2 (the tile-t+1 pair) guarantees tile t landed.
        if (t + 1 < TLEN) __builtin_amdgcn_s_wait_asynccnt(2);
        else              __builtin_amdgcn_s_wait_asynccnt(0);
#endif
        __syncthreads();

        // phase C: S = P x E^T via 16 chained f32 WMMAs (K = 64 in slices of 4)
        v8f acc = {};
        #pragma unroll
        for (int c = 0; c < 16; ++c) {
            v2f a = *(const v2f*)&pmat[arow * 64 + 4 * c + koff];
            acc = __builtin_amdgcn_wmma_f32_16x16x4_f32(
                      false, a, false, bfrag[c], (short)0, acc, false, false);
        }

        // phase D: alpha[b,i] = log(S) + m[b] + logit[b,t,i], masked by lens
        // D layout: lanes 0-15 -> N=lane, M=v; lanes 16-31 -> N=lane-16, M=8+v
        #pragma unroll
        for (int v = 0; v < 8; ++v) {
            int bi = ((lane >> 4) << 3) + v;          // batch row 0..15
            int ii = wave * 16 + (lane & 15);         // label index
            float na = fast_log(acc[v]) + mrow[bi] + ltc[bi * 64 + ii];
            if (t < lens_s[bi]) alpha[bi * 64 + ii] = na;
        }
        __syncthreads();
    }

    // ---- finalize: out[b] = logsumexp_i(alpha[b,i] + trans[stop,i])
    if (tid < 16) {
        float mx = -3.4e38f;
        for (int j = 0; j < 64; ++j) mx = fmaxf(mx, alpha[tid * 64 + j] + ts[j]);
        float s = 0.0f;
        for (int j = 0; j < 64; ++j) s += __expf(alpha[tid * 64 + j] + ts[j] - mx);
        out[b0 + tid] = mx + fast_log(s);
    }
}

extern "C" void kernel_launch(void* const* d_in, const int* in_sizes, int n_in,
                              void* d_out, int out_size, void* d_ws, size_t ws_size,
                              hipStream_t stream) {
    const float* logits = (const float*)d_in[0];
    const int*   lens   = (const int*)d_in[1];
    const float* trans  = (const float*)d_in[2];
    float* out = (float*)d_out;
    crf_fwd_kernel<<<BSZ / 16, 128, 0, stream>>>(logits, lens, trans, out);
}